// GraphSAGEEncoder_24421184045372
// MI455X (gfx1250) — compile-verified
//
#include <hip/hip_runtime.h>
#include <hip/hip_bf16.h>
#include <math.h>

typedef __attribute__((ext_vector_type(2))) float v2f;
typedef __attribute__((ext_vector_type(8))) float v8f;

#define D 128
#define ROWS_PER_BLK 16
#define LDS_STRIDE 132   // pad 128 -> 132 floats to spread LDS banks

// ---------------- edge kernels (memory-bound phase) ----------------

__global__ __launch_bounds__(256) void edge_count_kernel(
    const int* __restrict__ dst, int E, float* __restrict__ cnt) {
    int e = blockIdx.x * blockDim.x + threadIdx.x;
    if (e < E) atomicAdd(&cnt[dst[e]], 1.0f);
}

// one thread = one (edge, 4-float chunk); 32 threads cover one edge's 128 feats
__global__ __launch_bounds__(256) void edge_scatter_kernel(
    const float* __restrict__ x, const int* __restrict__ src,
    const int* __restrict__ dst, int E, float* __restrict__ agg) {
    long long gid = (long long)blockIdx.x * blockDim.x + threadIdx.x;
    int e = (int)(gid >> 5);
    if (e >= E) return;
    int c = ((int)gid & 31) * 4;
    const float4 v = *(const float4*)(x + (size_t)src[e] * D + c);
    float* a = agg + (size_t)dst[e] * D + c;
    atomicAdd(a + 0, v.x);
    atomicAdd(a + 1, v.y);
    atomicAdd(a + 2, v.z);
    atomicAdd(a + 3, v.w);
}

__global__ __launch_bounds__(256) void make_scale_kernel(
    float* __restrict__ cnt, int n) {
    int i = blockIdx.x * blockDim.x + threadIdx.x;
    if (i < n) cnt[i] = 1.0f / fmaxf(cnt[i], 1.0f);
}

// ---------------- fused SAGE GEMM: out = (agg*scale)@Wl + x@Wr + bl ----------------
// block = 256 threads = 8 waves; block computes 16 rows x 128 cols.
// wave w computes the 16x16 tile at cols [16w, 16w+16) via V_WMMA_F32_16X16X4_F32.

template <bool APPLY_GELU>
__global__ __launch_bounds__(256) void sage_gemm_kernel(
    const float* __restrict__ x,      // [N,128] self features
    const float* __restrict__ agg,    // [N,128] summed neighbor features
    const float* __restrict__ scale,  // [N] 1/max(cnt,1)
    const float* __restrict__ Wl,     // [128,128]
    const float* __restrict__ Wr,     // [128,128]
    const float* __restrict__ bl,     // [128]
    float* __restrict__ out,          // [N,128]
    int N) {
    __shared__ float sX[ROWS_PER_BLK][LDS_STRIDE];
    __shared__ float sM[ROWS_PER_BLK][LDS_STRIDE];
    __shared__ float sScale[ROWS_PER_BLK];

    const int tid = threadIdx.x;
    const int lane = tid & 31;
    const int wave = tid >> 5;
    const int rowBase = blockIdx.x * ROWS_PER_BLK;

    if (tid < ROWS_PER_BLK) {
        int gr = rowBase + tid;
        if (gr >= N) gr = N - 1;
        sScale[tid] = scale[gr];
    }
    // stage this block's 16x128 x-tile and agg-tile into LDS (shared by 8 waves)
    for (int t = tid; t < ROWS_PER_BLK * D; t += 256) {
        int rr = t >> 7;          // / 128
        int cc = t & (D - 1);
        int gr = rowBase + rr;
        if (gr >= N) gr = N - 1;
        sX[rr][cc] = x[(size_t)gr * D + cc];
        sM[rr][cc] = agg[(size_t)gr * D + cc];
    }
    __syncthreads();

    // A-fragment lane mapping (16x4 f32): row = lane&15, K-pair = (lane>>4)*2
    // B-fragment lane mapping (4x16 f32): col = lane&15, K-pair = (lane>>4)*2
    const int r = lane & 15;
    const int kh = (lane >> 4) * 2;
    const int col = wave * 16 + r;
    const float sc = sScale[r];

    v8f acc = {0.f, 0.f, 0.f, 0.f, 0.f, 0.f, 0.f, 0.f};

    for (int k0 = 0; k0 < D; k0 += 4) {
        const int k = k0 + kh;
        // mean @ Wl  (scale folded into A)
        v2f am = *(const v2f*)&sM[r][k];
        am.x *= sc;
        am.y *= sc;
        v2f blw;
        blw.x = Wl[(size_t)k * D + col];
        blw.y = Wl[(size_t)(k + 1) * D + col];
        acc = __builtin_amdgcn_wmma_f32_16x16x4_f32(
            false, am, false, blw, (short)0, acc, false, false);
        // x @ Wr
        v2f ax = *(const v2f*)&sX[r][k];
        v2f brw;
        brw.x = Wr[(size_t)k * D + col];
        brw.y = Wr[(size_t)(k + 1) * D + col];
        acc = __builtin_amdgcn_wmma_f32_16x16x4_f32(
            false, ax, false, brw, (short)0, acc, false, false);
    }

    const float bias = bl[col];
    // C/D layout: VGPR v -> M = v + 8*(lane>>4), N = lane&15
    const int mOff = (lane >> 4) * 8;
#pragma unroll
    for (int v = 0; v < 8; ++v) {
        float o = acc[v] + bias;
        if (APPLY_GELU) {
            o = 0.5f * o * (1.0f + erff(o * 0.70710678118654752f));
        }
        const int orow = rowBase + v + mOff;
        if (orow < N) out[(size_t)orow * D + col] = o;
    }
}

// ---------------- host-side orchestration ----------------

static void run_layer(const float* x, const int* ei, int E,
                      const float* Wl, const float* bl, const float* Wr,
                      float* out, float* agg, float* cnt, int N,
                      bool gelu, hipStream_t stream) {
    const int* src = ei;
    const int* dst = ei + E;
    hipMemsetAsync(agg, 0, ((size_t)N * D + N) * sizeof(float), stream);
    edge_count_kernel<<<(E + 255) / 256, 256, 0, stream>>>(dst, E, cnt);
    long long sthreads = (long long)E * 32;
    edge_scatter_kernel<<<(unsigned)((sthreads + 255) / 256), 256, 0, stream>>>(
        x, src, dst, E, agg);
    make_scale_kernel<<<(N + 255) / 256, 256, 0, stream>>>(cnt, N);
    int blocks = (N + ROWS_PER_BLK - 1) / ROWS_PER_BLK;
    if (gelu) {
        sage_gemm_kernel<true><<<blocks, 256, 0, stream>>>(
            x, agg, cnt, Wl, Wr, bl, out, N);
    } else {
        sage_gemm_kernel<false><<<blocks, 256, 0, stream>>>(
            x, agg, cnt, Wl, Wr, bl, out, N);
    }
}

extern "C" void kernel_launch(void* const* d_in, const int* in_sizes, int n_in,
                              void* d_out, int out_size, void* d_ws, size_t ws_size,
                              hipStream_t stream) {
    const float* embs = (const float*)d_in[0];
    const int* ei0 = (const int*)d_in[1];
    const int* ei1 = (const int*)d_in[2];
    const float* Wl0 = (const float*)d_in[3];
    const float* bl0 = (const float*)d_in[4];
    const float* Wr0 = (const float*)d_in[5];
    const float* Wl1 = (const float*)d_in[6];
    const float* bl1 = (const float*)d_in[7];
    const float* Wr1 = (const float*)d_in[8];

    const int N = in_sizes[0] / D;
    const int E0 = in_sizes[1] / 2;
    const int E1 = in_sizes[2] / 2;

    float* agg = (float*)d_ws;                 // [N,128]
    float* cnt = agg + (size_t)N * D;          // [N]  (reused as scale)
    float* out = (float*)d_out;                // [N,128]; also holds h between layers

    // layer 0: h = gelu(mean0 @ Wl0 + bl0 + embs @ Wr0)  -> stored in d_out
    run_layer(embs, ei0, E0, Wl0, bl0, Wr0, out, agg, cnt, N, true, stream);
    // layer 1: out = mean1 @ Wl1 + bl1 + h @ Wr1 (in place on d_out; reads of a
    // block's own rows complete at the staging barrier before any writes)
    run_layer(out, ei1, E1, Wl1, bl1, Wr1, out, agg, cnt, N, false, stream);
}